// CrossNet_78022375899168
// MI455X (gfx1250) — compile-verified
//
#include <hip/hip_runtime.h>

// CrossNet (DCN-v1 style) fused for MI455X / gfx1250.
// B=16384 rows, D=1024 cols, 4 layers, BatchNorm over the batch dim each layer.
//
// Strategy (memory-bound: ~0.5 GFLOP vs ~768MB minimal traffic, L2-resident):
//  - cross_fused: one wave owns one row (32 f32/lane in VGPRs). Applies the
//    previous layer's BN, computes s = row.w via wave32 butterfly, does
//    res = res + x0*s + b, accumulates per-column sum/sum^2 in registers,
//    writes the row. Exactly 1 read of res, 1 read of x0, 1 write per layer.
//    Per-block column partials are merged in LDS with a deterministic
//    sequential-wave scheme (no float atomics) and written to workspace.
//  - cross_stats: cross-block column reduction done with V_WMMA_F32_16X16X4_F32
//    (A = ones 16x4, B = 4x16 tile of partials -> D rows are the column sums).
//  - cross_norm: final BN apply, in place on d_out.

#define B_ROWS   16384
#define D_DIM    1024
#define N_LAYER  4
#define BN_EPS   1e-5f

#define NBLK           256   // blocks in cross_fused
#define WAVES_PER_BLK  8     // 256 threads, wave32
#define ROWS_PER_WAVE  (B_ROWS / (NBLK * WAVES_PER_BLK))   // 8

typedef __attribute__((ext_vector_type(2))) float v2f;
typedef __attribute__((ext_vector_type(8))) float v8f;

// ---------------------------------------------------------------------------
// Fused layer kernel: BN(prev) -> dot -> update -> stat partials
// ---------------------------------------------------------------------------
__global__ __launch_bounds__(256) void cross_fused(
    const float* __restrict__ resin,   // (B,D) raw res of prev layer (or x for l=0)
    const float* __restrict__ x0,      // (B,D) original x
    const float* __restrict__ w,       // (D,) this layer's weight
    const float* __restrict__ bvec,    // (D,) this layer's bias
    const float* __restrict__ stats,   // mu[0..D) , rsigma[D..2D)  (prev layer)
    float* __restrict__ resout,        // (B,D) raw res of this layer
    float* __restrict__ partials,      // (NBLK, 2, D) per-block col sums / sumsq
    int apply_prev)
{
    __shared__ float4 sW[D_DIM / 4], sB[D_DIM / 4], sMu[D_DIM / 4], sRs[D_DIM / 4];
    __shared__ float4 sAccS[D_DIM / 4], sAccQ[D_DIM / 4];

    const int tid = threadIdx.x;           // 0..255, one float4 column chunk each
    {
        const float4* w4  = (const float4*)w;
        const float4* b4  = (const float4*)bvec;
        sW[tid] = w4[tid];
        sB[tid] = b4[tid];
        if (apply_prev) {
            const float4* mu4 = (const float4*)stats;
            const float4* rs4 = (const float4*)(stats + D_DIM);
            sMu[tid] = mu4[tid];
            sRs[tid] = rs4[tid];
        }
        sAccS[tid] = make_float4(0.f, 0.f, 0.f, 0.f);
        sAccQ[tid] = make_float4(0.f, 0.f, 0.f, 0.f);
    }
    __syncthreads();

    const int wave = tid >> 5;
    const int lane = tid & 31;
    const int rowBase = (blockIdx.x * WAVES_PER_BLK + wave) * ROWS_PER_WAVE;

    float4 accS[8], accQ[8];
#pragma unroll
    for (int j = 0; j < 8; ++j) {
        accS[j] = make_float4(0.f, 0.f, 0.f, 0.f);
        accQ[j] = make_float4(0.f, 0.f, 0.f, 0.f);
    }

    for (int r = 0; r < ROWS_PER_WAVE; ++r) {
        const int row = rowBase + r;
        const float4* rp = (const float4*)(resin + (size_t)row * D_DIM);
        const float4* xp = (const float4*)(x0    + (size_t)row * D_DIM);

        float4 rv[8], xv[8];
#pragma unroll
        for (int j = 0; j < 8; ++j) {
            const int fi = lane + 32 * j;   // coalesced: lane-consecutive float4
            rv[j] = rp[fi];
            xv[j] = xp[fi];
        }

        if (apply_prev) {
#pragma unroll
            for (int j = 0; j < 8; ++j) {
                const int fi = lane + 32 * j;
                const float4 m = sMu[fi], s = sRs[fi];
                rv[j].x = (rv[j].x - m.x) * s.x;
                rv[j].y = (rv[j].y - m.y) * s.y;
                rv[j].z = (rv[j].z - m.z) * s.z;
                rv[j].w = (rv[j].w - m.w) * s.w;
            }
        }

        // s = dot(row, w): per-lane partial then wave32 butterfly reduce
        float p = 0.f;
#pragma unroll
        for (int j = 0; j < 8; ++j) {
            const int fi = lane + 32 * j;
            const float4 ww = sW[fi];
            p = fmaf(rv[j].x, ww.x, p);
            p = fmaf(rv[j].y, ww.y, p);
            p = fmaf(rv[j].z, ww.z, p);
            p = fmaf(rv[j].w, ww.w, p);
        }
#pragma unroll
        for (int off = 16; off > 0; off >>= 1)
            p += __shfl_xor(p, off, 32);    // all lanes now hold s

        float4* op = (float4*)(resout + (size_t)row * D_DIM);
#pragma unroll
        for (int j = 0; j < 8; ++j) {
            const int fi = lane + 32 * j;
            const float4 bb = sB[fi];
            float4 o;
            o.x = fmaf(xv[j].x, p, rv[j].x) + bb.x;
            o.y = fmaf(xv[j].y, p, rv[j].y) + bb.y;
            o.z = fmaf(xv[j].z, p, rv[j].z) + bb.z;
            o.w = fmaf(xv[j].w, p, rv[j].w) + bb.w;
            accS[j].x += o.x;  accQ[j].x = fmaf(o.x, o.x, accQ[j].x);
            accS[j].y += o.y;  accQ[j].y = fmaf(o.y, o.y, accQ[j].y);
            accS[j].z += o.z;  accQ[j].z = fmaf(o.z, o.z, accQ[j].z);
            accS[j].w += o.w;  accQ[j].w = fmaf(o.w, o.w, accQ[j].w);
            op[fi] = o;
        }
    }

    // Deterministic cross-wave merge of column accumulators (no float atomics)
    for (int wv = 0; wv < WAVES_PER_BLK; ++wv) {
        if (wave == wv) {
#pragma unroll
            for (int j = 0; j < 8; ++j) {
                const int fi = lane + 32 * j;
                float4 t = sAccS[fi];
                t.x += accS[j].x; t.y += accS[j].y; t.z += accS[j].z; t.w += accS[j].w;
                sAccS[fi] = t;
                float4 q = sAccQ[fi];
                q.x += accQ[j].x; q.y += accQ[j].y; q.z += accQ[j].z; q.w += accQ[j].w;
                sAccQ[fi] = q;
            }
        }
        __syncthreads();
    }

    // Write block partials: [blk][0][D] = colsum, [blk][1][D] = colsumsq
    float4* ps = (float4*)(partials + (size_t)blockIdx.x * 2 * D_DIM);
    ps[tid]                 = sAccS[tid];
    ps[(D_DIM / 4) + tid]   = sAccQ[tid];
}

// ---------------------------------------------------------------------------
// Cross-block column reduction via V_WMMA_F32_16X16X4_F32.
// Partials P viewed as NBLK x 2048 row-major (cols 0..1023 = sum, +1024 = sq).
// A = ones(16x4)  =>  every row of D = column sums of B(4x16); accumulate K.
// One 32-thread block (one wave) -> 16 columns' (mu, rsigma).
// ---------------------------------------------------------------------------
__global__ __launch_bounds__(32) void cross_stats(
    const float* __restrict__ partials,   // (NBLK, 2*D)
    float* __restrict__ stats)            // mu[0..D), rsigma[D..2D)
{
    const int lane    = threadIdx.x;
    const int colbase = blockIdx.x * 16;
    const int n       = lane & 15;
    const int koff    = (lane < 16) ? 0 : 2;   // ISA 32-bit A/B layout: K 0/2 by lane half

    v8f cS = {0.f, 0.f, 0.f, 0.f, 0.f, 0.f, 0.f, 0.f};
    v8f cQ = {0.f, 0.f, 0.f, 0.f, 0.f, 0.f, 0.f, 0.f};
    const v2f a = {1.0f, 1.0f};                // A = ones for all (m,k)

    const size_t rstride = (size_t)2 * D_DIM;
    for (int kb = 0; kb < NBLK; kb += 4) {
        const float* b0 = partials + (size_t)(kb + koff) * rstride + colbase + n;
        const float* b1 = b0 + rstride;        // next K row
        v2f bS; bS[0] = b0[0];      bS[1] = b1[0];
        v2f bQ; bQ[0] = b0[D_DIM];  bQ[1] = b1[D_DIM];
        cS = __builtin_amdgcn_wmma_f32_16x16x4_f32(
                 false, a, false, bS, (short)0, cS, false, false);
        cQ = __builtin_amdgcn_wmma_f32_16x16x4_f32(
                 false, a, false, bQ, (short)0, cQ, false, false);
    }

    // D row M=0 lives in c[0] on lanes 0..15 (N = lane)
    const float s = cS[0];
    const float q = cQ[0];
    if (lane < 16) {
        const int col  = colbase + lane;
        const float mu  = s * (1.0f / (float)B_ROWS);
        const float var = q * (1.0f / (float)B_ROWS) - mu * mu;  // biased var
        stats[col]         = mu;
        stats[D_DIM + col] = rsqrtf(var + BN_EPS);
    }
}

// ---------------------------------------------------------------------------
// Final BN apply, in place on d_out.
// ---------------------------------------------------------------------------
__global__ __launch_bounds__(256) void cross_norm(
    float* __restrict__ res, const float* __restrict__ stats)
{
    const size_t i = (size_t)blockIdx.x * blockDim.x + threadIdx.x;  // float4 idx
    const int col4 = (int)(i & (D_DIM / 4 - 1));
    const float4 m  = ((const float4*)stats)[col4];
    const float4 rs = ((const float4*)(stats + D_DIM))[col4];
    float4* p = (float4*)res;
    float4 v = p[i];
    v.x = (v.x - m.x) * rs.x;
    v.y = (v.y - m.y) * rs.y;
    v.z = (v.z - m.z) * rs.z;
    v.w = (v.w - m.w) * rs.w;
    p[i] = v;
}

// ---------------------------------------------------------------------------
extern "C" void kernel_launch(void* const* d_in, const int* in_sizes, int n_in,
                              void* d_out, int out_size, void* d_ws, size_t ws_size,
                              hipStream_t stream) {
    (void)in_sizes; (void)n_in; (void)out_size; (void)ws_size;

    const float* x  = (const float*)d_in[0];   // (B,D)
    const float* ww = (const float*)d_in[1];   // (4,D)
    const float* wb = (const float*)d_in[2];   // (4,D)

    float* res      = (float*)d_out;                            // res buffer (in place)
    float* partials = (float*)d_ws;                             // NBLK*2*D floats (2MB)
    float* stats    = partials + (size_t)NBLK * 2 * D_DIM;      // 2*D floats

    for (int l = 0; l < N_LAYER; ++l) {
        cross_fused<<<NBLK, 256, 0, stream>>>(
            (l == 0) ? x : res, x, ww + (size_t)l * D_DIM, wb + (size_t)l * D_DIM,
            stats, res, partials, (l == 0) ? 0 : 1);
        cross_stats<<<D_DIM / 16, 32, 0, stream>>>(partials, stats);
    }
    const int norm_blocks = (int)(((size_t)B_ROWS * D_DIM / 4) / 256);  // 16384
    cross_norm<<<norm_blocks, 256, 0, stream>>>(res, stats);
}